// ValueTimeEmbedding_53154515255328
// MI455X (gfx1250) — compile-verified
//
#include <hip/hip_runtime.h>
#include <math.h>

// Problem constants (from reference)
#define FF    64
#define HH    8
#define EE    64
#define PP    512
#define BT    (64 * 256)     // 16384 rows
#define KTOT  1024           // 2 branches * F * H

typedef float v2f __attribute__((ext_vector_type(2)));
typedef float v4f __attribute__((ext_vector_type(4)));
typedef float v8f __attribute__((ext_vector_type(8)));
typedef int   v4i __attribute__((ext_vector_type(4)));

// --------------------------------------------------------------------------
// CDNA5 async global->LDS path (ASYNCcnt), guarded so the file compiles on
// toolchains without the builtins (falls back to VGPR round-trip staging).
// Builtin signatures (from clang diagnostics):
//   b128: (AS1 int4* src, AS3 int4* dst, imm int offset, imm int cpol)
//   b32:  (AS1 int*  src, AS3 int*  dst, imm int offset, imm int cpol)
// --------------------------------------------------------------------------
#if defined(__gfx1250__) && \
    __has_builtin(__builtin_amdgcn_global_load_async_to_lds_b128) && \
    __has_builtin(__builtin_amdgcn_global_load_async_to_lds_b32)
#define USE_ASYNC 1
#else
#define USE_ASYNC 0
#endif

#if USE_ASYNC
#if __has_builtin(__builtin_amdgcn_s_wait_asynccnt)
#define WAIT_ASYNC(n) __builtin_amdgcn_s_wait_asynccnt(n)
#else
#define WAIT_ASYNC(n) asm volatile("s_wait_asynccnt " #n ::: "memory")
#endif
#else
#define WAIT_ASYNC(n) do {} while (0)
#endif

#if USE_ASYNC
typedef __attribute__((address_space(1))) v4i  g_v4i;
typedef __attribute__((address_space(3))) v4i  l_v4i;
typedef __attribute__((address_space(1))) int  g_i32;
typedef __attribute__((address_space(3))) int  l_i32;
#endif

__device__ __forceinline__ void async_copy_b128(const float* g, float* l) {
#if USE_ASYNC
  __builtin_amdgcn_global_load_async_to_lds_b128((g_v4i*)g, (l_v4i*)l, 0, 0);
#else
  *(v4f*)l = *(const v4f*)g;
#endif
}
__device__ __forceinline__ void async_copy_b32(const float* g, float* l) {
#if USE_ASYNC
  __builtin_amdgcn_global_load_async_to_lds_b32((g_i32*)g, (l_i32*)l, 0, 0);
#else
  *l = *g;
#endif
}

// ---------------------------------------------------------------------------
// Kernel 1: fold w2 (H->E) into wx/wt (E->P):
//   Wcat[k][p], k = branch*512 + f*8 + h, = sum_e w2[f][h][e] * W[(f*E+e)*P+p]
// ---------------------------------------------------------------------------
__global__ __launch_bounds__(256) void build_w_kernel(
    const float* __restrict__ w2v, const float* __restrict__ w2t,
    const float* __restrict__ wx,  const float* __restrict__ wt,
    float* __restrict__ Wcat) {
  int idx = blockIdx.x * blockDim.x + threadIdx.x;   // [0, KTOT*PP)
  int p  = idx & (PP - 1);
  int k  = idx >> 9;
  int br = k >> 9;
  int kk = k & 511;
  int f  = kk >> 3;
  int h  = kk & 7;
  const float* w2 = br ? w2t : w2v;
  const float* wb = br ? wt  : wx;
  const float* w2row = w2 + (size_t)(f * HH + h) * EE;
  const float* wcol  = wb + (size_t)f * EE * PP + p;
  float acc = 0.f;
#pragma unroll 8
  for (int e = 0; e < EE; ++e)
    acc = fmaf(w2row[e], wcol[(size_t)e * PP], acc);
  Wcat[idx] = acc;
}

// ---------------------------------------------------------------------------
// Kernel 2: tanh prologue, materialize Hcat[16384][1024] (L2-resident, 64 MB)
// ---------------------------------------------------------------------------
__global__ __launch_bounds__(256) void build_h_kernel(
    const float* __restrict__ x,   const float* __restrict__ tm,
    const float* __restrict__ w1v, const float* __restrict__ b1v,
    const float* __restrict__ w1t, const float* __restrict__ b1t,
    float* __restrict__ Hcat) {
  int idx = blockIdx.x * blockDim.x + threadIdx.x;   // [0, BT*KTOT)
  int k  = idx & (KTOT - 1);
  int bt = idx >> 10;
  int br = k >> 9;
  int kk = k & 511;
  int f  = kk >> 3;
  int h  = kk & 7;
  const float* in = br ? tm  : x;
  const float* w1 = br ? w1t : w1v;
  const float* b1 = br ? b1t : b1v;
  Hcat[idx] = tanhf(fmaf(in[bt * FF + f], w1[f * HH + h], b1[f * HH + h]));
}

// ---------------------------------------------------------------------------
// Kernel 3: C = Hcat[16384][1024] @ Wcat[1024][512] + (bx + bt)
//
// Block tile 128(M) x 64(N), 256 threads = 8 waves; wave w owns rows w*16..+15
// as four 16x16 f32 tiles via V_WMMA_F32_16X16X4_F32 (exact fp32).
//
// Double-buffered LDS, filled by GLOBAL_LOAD_ASYNC_TO_LDS (ASYNCcnt) so the
// HBM/L2 fetch of tile i+1 overlaps the WMMA stream of tile i.
//   A: row-major, stride 36 floats (144B rows: 16B aligned; gcd(36,64)=4 gives
//      16 distinct banks for the frag reads, halves on even/odd bank pairs).
//   B: k-pair interleaved, (k,n) at [(k>>1)*160 + 2n + (k&1)] so each lane's
//      (b.x,b.y) is one aligned ds_load_b64; +160 floats = +32 banks keeps the
//      two lane-halves on disjoint bank sets.
// ---------------------------------------------------------------------------
#define MB     128
#define NB     64
#define KB     32
#define LDA    36
#define PAIRS  160           // floats per k-pair group of B (2*NB + 32 pad)

__global__ __launch_bounds__(256) void gemm_kernel(
    const float* __restrict__ A,    // Hcat [BT][KTOT]
    const float* __restrict__ Bm,   // Wcat [KTOT][PP]
    const float* __restrict__ bx,   const float* __restrict__ btb,
    float* __restrict__ C) {
  __shared__ __align__(16) float lsA[2][MB * LDA];          // 2 x 18 KB
  __shared__ __align__(16) float lsB[2][(KB / 2) * PAIRS];  // 2 x 10 KB

  const int tid  = threadIdx.x;
  const int lane = tid & 31;
  const int wave = tid >> 5;
  const int row0 = blockIdx.x * MB;
  const int col0 = blockIdx.y * NB;

  const int r16  = lane & 15;       // 0..15
  const int koff = (lane >> 4) * 2; // 0 for lanes 0-15, 2 for lanes 16-31

  v8f acc[4] = {};

  // Issue one tile stage: 4x b128 (A) + 8x b32 (B) per thread
  // = 12 async instructions per wave.
  auto issue_stage = [&](int buf, int kb) {
#pragma unroll
    for (int pass = 0; pass < 4; ++pass) {      // A: 128x32 floats
      int g  = pass * 256 + tid;                // 0..1023
      int r  = g >> 3;                          // row 0..127
      int c4 = (g & 7) << 2;                    // col 0,4,...,28
      async_copy_b128(&A[(size_t)(row0 + r) * KTOT + kb + c4],
                      &lsA[buf][r * LDA + c4]);
    }
#pragma unroll
    for (int pass = 0; pass < 8; ++pass) {      // B: 32x64 floats, coalesced
      int g  = pass * 256 + tid;                // 0..2047
      int rr = g >> 6;                          // k row 0..31
      int n  = g & 63;                          // col 0..63
      async_copy_b32(&Bm[(size_t)(kb + rr) * PP + col0 + n],
                     &lsB[buf][(rr >> 1) * PAIRS + n * 2 + (rr & 1)]);
    }
  };

  issue_stage(0, 0);

  const int NKB = KTOT / KB;                    // 32 k-tiles
  for (int it = 0; it < NKB; ++it) {
    const int buf = it & 1;
    if (it + 1 < NKB) {
      issue_stage(buf ^ 1, (it + 1) * KB);      // prefetch next tile
      WAIT_ASYNC(12);                           // current tile's 12 done
    } else {
      WAIT_ASYNC(0);
    }
    __syncthreads();

    const float* aRow = &lsA[buf][(wave * 16 + r16) * LDA];
    const float* bBuf = &lsB[buf][0];
#pragma unroll
    for (int kk = 0; kk < KB; kk += 4) {
      v2f a = *(const v2f*)&aRow[kk + koff];                 // ds_load_b64
      const float* bRow = &bBuf[((kk + koff) >> 1) * PAIRS];
#pragma unroll
      for (int t = 0; t < 4; ++t) {
        v2f b = *(const v2f*)&bRow[(t * 16 + r16) * 2];      // ds_load_b64
        acc[t] = __builtin_amdgcn_wmma_f32_16x16x4_f32(
            false, a, false, b, (short)0, acc[t], false, false);
      }
    }
    __syncthreads();   // all waves done reading buf before it is refilled
  }

  // --- epilogue: bias add + store (ISA C/D layout: lane half -> M +8) ---
  const int half = lane >> 4;
#pragma unroll
  for (int t = 0; t < 4; ++t) {
    int gcol = col0 + t * 16 + r16;
    float bias = bx[gcol] + btb[gcol];
#pragma unroll
    for (int i = 0; i < 8; ++i) {
      int grow = row0 + wave * 16 + half * 8 + i;
      C[(size_t)grow * PP + gcol] = acc[t][i] + bias;
    }
  }
}

// ---------------------------------------------------------------------------
extern "C" void kernel_launch(void* const* d_in, const int* in_sizes, int n_in,
                              void* d_out, int out_size, void* d_ws, size_t ws_size,
                              hipStream_t stream) {
  const float* x   = (const float*)d_in[0];
  const float* tm  = (const float*)d_in[1];
  const float* w1v = (const float*)d_in[2];
  const float* b1v = (const float*)d_in[3];
  const float* w2v = (const float*)d_in[4];
  const float* w1t = (const float*)d_in[5];
  const float* b1t = (const float*)d_in[6];
  const float* w2t = (const float*)d_in[7];
  const float* wx  = (const float*)d_in[8];
  const float* bx  = (const float*)d_in[9];
  const float* wt  = (const float*)d_in[10];
  const float* bt  = (const float*)d_in[11];
  float* out = (float*)d_out;

  float* Wcat = (float*)d_ws;                          // 1024*512 f32 = 2 MB
  float* Hcat = Wcat + (size_t)KTOT * PP;              // 16384*1024 f32 = 64 MB

  build_w_kernel<<<(KTOT * PP) / 256, 256, 0, stream>>>(w2v, w2t, wx, wt, Wcat);
  build_h_kernel<<<(BT * KTOT) / 256, 256, 0, stream>>>(x, tm, w1v, b1v, w1t, b1t, Hcat);

  dim3 grid(BT / MB, PP / NB);  // 128 x 8
  gemm_kernel<<<grid, 256, 0, stream>>>(Hcat, Wcat, bx, bt, out);
}